// SparseCodingLoss_81664508166413
// MI455X (gfx1250) — compile-verified
//
#include <hip/hip_runtime.h>
#include <stdint.h>
#include <math.h>

// ---------------- problem constants ----------------
#define S_STEPS   64
#define BATCH     4
#define ATOMS     256
#define TIME_T    1024
#define ATOM_EMB  190
#define EMB_DIM   256
#define AT_ELEMS  (ATOMS * TIME_T)        // 262144 elems per (s,b) feature map
#define PARTS     4                       // split each argmax job 4 ways
#define PART_ELEMS (AT_ELEMS / PARTS)     // 65536
#define P1_THREADS 256
#define TILE_ELEMS (P1_THREADS * 4)       // 1024 floats per LDS tile
#define TILES      (PART_ELEMS / TILE_ELEMS) // 64 tiles per block
#define ASYNC_DEPTH 4                     // async pipeline depth (tiles in flight)
#define N_JOBS    (2 * S_STEPS * BATCH)   // 512

// ---------------- CDNA5 async-tensor helpers ----------------
// GV-form async copy: 16B/lane from global -> LDS, tracked by ASYNCcnt.
__device__ __forceinline__ void async_copy_b128(uint32_t lds_off, const void* gptr) {
    asm volatile("global_load_async_to_lds_b128 %0, %1, off"
                 :: "v"(lds_off), "v"((uint64_t)(uintptr_t)gptr)
                 : "memory");
}
template <int N>
__device__ __forceinline__ void wait_asynccnt_le() {
    asm volatile("s_wait_asynccnt %0" :: "n"(N) : "memory");
}
__device__ __forceinline__ void wait_async_dyn(int n) {
    switch (n) {
        case 0:  wait_asynccnt_le<0>(); break;
        case 1:  wait_asynccnt_le<1>(); break;
        case 2:  wait_asynccnt_le<2>(); break;
        default: wait_asynccnt_le<3>(); break;
    }
}
__device__ __forceinline__ void wait_dscnt0() {
    // guarantees prior LDS reads completed before we let the async engine
    // overwrite the same tile buffer (WAR on LDS)
    asm volatile("s_wait_dscnt 0" ::: "memory");
}

// ---------------- phase 1: streaming argmax (HBM-bound, async pipeline) ----------------
__global__ __launch_bounds__(P1_THREADS)
void phase1_argmax(const float* __restrict__ a_fms, const float* __restrict__ b_fms,
                   float* __restrict__ pvals, int* __restrict__ pidxs) {
    __shared__ float4 buf[ASYNC_DEPTH][P1_THREADS];   // 16 KB tile ring
    __shared__ float  rv[P1_THREADS];
    __shared__ int    ri[P1_THREADS];

    const int tid  = threadIdx.x;
    const int blk  = blockIdx.x;            // 0..2047
    const int part = blk & (PARTS - 1);
    const int job  = blk >> 2;              // 0..511 : tensor*256 + (s*4 + b)
    const int tsr  = job >> 8;
    const int sb   = job & 255;

    const float* src = (tsr == 0 ? a_fms : b_fms);
    src += (size_t)sb * AT_ELEMS + (size_t)part * PART_ELEMS;
    const char* gbase = (const char*)src + (size_t)tid * 16;

    uint32_t lds[ASYNC_DEPTH];
#pragma unroll
    for (int d = 0; d < ASYNC_DEPTH; ++d)
        lds[d] = (uint32_t)(uintptr_t)(const void*)&buf[d][tid];

    // prologue: fill the pipeline
#pragma unroll
    for (int k = 0; k < ASYNC_DEPTH; ++k)
        async_copy_b128(lds[k], gbase + (size_t)k * (TILE_ELEMS * 4));

    float best = -INFINITY;
    int   bidx = 0;
    for (int k = 0; k < TILES; ++k) {
        const int issued = (k + ASYNC_DEPTH < TILES) ? (k + ASYNC_DEPTH) : TILES;
        wait_async_dyn(issued - (k + 1));           // tile k complete
        float4 v = buf[k & (ASYNC_DEPTH - 1)][tid];
        const int fb = part * PART_ELEMS + k * TILE_ELEMS + tid * 4;
        if (v.x > best) { best = v.x; bidx = fb; }
        if (v.y > best) { best = v.y; bidx = fb + 1; }
        if (v.z > best) { best = v.z; bidx = fb + 2; }
        if (v.w > best) { best = v.w; bidx = fb + 3; }
        wait_dscnt0();                              // read done -> safe to reuse buffer
        if (k + ASYNC_DEPTH < TILES)
            async_copy_b128(lds[k & (ASYNC_DEPTH - 1)],
                            gbase + (size_t)(k + ASYNC_DEPTH) * (TILE_ELEMS * 4));
    }

    // block reduce: max with lowest-index tie-break (jnp.argmax semantics)
    rv[tid] = best;
    ri[tid] = bidx;
    __syncthreads();
    for (int s = P1_THREADS / 2; s > 0; s >>= 1) {
        if (tid < s) {
            float ov = rv[tid + s]; int oi = ri[tid + s];
            if (ov > rv[tid] || (ov == rv[tid] && oi < ri[tid])) { rv[tid] = ov; ri[tid] = oi; }
        }
        __syncthreads();
    }
    if (tid == 0) { pvals[blk] = rv[0]; pidxs[blk] = ri[0]; }
}

// ---------------- phase 2: per-job embedding + score ----------------
__device__ __forceinline__ float pos_encode_elem(float x, int j) {
    // [x, sin(2^0 x), cos(2^0 x), ..., sin(2^15 x), cos(2^15 x)]
    if (j == 0) return x;
    const int i = (j - 1) >> 1;
    const float f = (float)(1 << i);
    return (((j - 1) & 1) == 0) ? sinf(f * x) : cosf(f * x);
}

__global__ __launch_bounds__(EMB_DIM)
void phase2_embed(const float* __restrict__ pvals, const int* __restrict__ pidxs,
                  const float* __restrict__ table, const float* __restrict__ w,
                  float* __restrict__ embs, float* __restrict__ scores) {
    __shared__ float red[EMB_DIM];
    const int job = blockIdx.x;             // 0..511
    const int tsr = job >> 8;
    const int sb  = job & 255;
    const int s   = sb >> 2;
    const int b   = sb & 3;

    // combine the 4 partial argmaxes (lowest index wins ties)
    float best = pvals[job * PARTS];
    int   bidx = pidxs[job * PARTS];
    for (int p = 1; p < PARTS; ++p) {
        float v = pvals[job * PARTS + p];
        int   i = pidxs[job * PARTS + p];
        if (v > best || (v == best && i < bidx)) { best = v; bidx = i; }
    }
    const int astar = bidx >> 10;           // / TIME_T
    const int tstar = bidx & (TIME_T - 1);

    // soft_dirac forward = argmax of the sparse sum vector (one nonzero = best)
    int a_eff, t_eff;
    if (best > 0.0f)       { a_eff = astar;                 t_eff = tstar; }
    else if (best == 0.0f) { a_eff = 0;                     t_eff = 0; }
    else                   { a_eff = (astar == 0) ? 1 : 0;  t_eff = (tstar == 0) ? 1 : 0; }

    const float spos = (float)t_eff * (1.0f / (float)(TIME_T - 1));

    const int e = threadIdx.x;              // 0..255
    float val;
    if (e < 33)       val = pos_encode_elem(spos, e);
    else if (e < 66)  val = pos_encode_elem(best, e - 33);
    else              val = table[a_eff * ATOM_EMB + (e - 66)];   // one-hot @ table == gather

    const int row = (tsr * BATCH + b) * S_STEPS + s;
    embs[row * EMB_DIM + e] = val;

    // score = dot(emb, order_w)
    red[e] = val * w[e];
    __syncthreads();
    for (int st = EMB_DIM / 2; st > 0; st >>= 1) {
        if (e < st) red[e] += red[e + st];
        __syncthreads();
    }
    if (e == 0) scores[row] = red[0];
}

// ---------------- phase 3: stable rank-sort + MSE ----------------
__global__ __launch_bounds__(256)
void phase3_loss(const float* __restrict__ embs, const float* __restrict__ scores,
                 float* __restrict__ out) {
    __shared__ float sc[2 * BATCH * S_STEPS];   // 512 scores
    __shared__ int   ord[2 * BATCH * S_STEPS];  // order[rank] -> step
    __shared__ float red[256];
    const int tid = threadIdx.x;

    sc[tid]       = scores[tid];
    sc[tid + 256] = scores[tid + 256];
    __syncthreads();

    // stable ascending rank (matches jnp.argsort): rank = #{s2 : key(s2) < key(s)}
    for (int g = tid; g < 2 * BATCH * S_STEPS; g += 256) {
        const int grp = g >> 6;                 // (tensor*4 + b), 0..7
        const int s   = g & 63;
        const float my = sc[g];
        int r = 0;
        for (int s2 = 0; s2 < S_STEPS; ++s2) {
            const float o = sc[(grp << 6) + s2];
            r += (o < my) || (o == my && s2 < s);
        }
        ord[(grp << 6) + r] = s;
    }
    __syncthreads();

    float acc = 0.0f;
    for (int j = tid; j < BATCH * S_STEPS * EMB_DIM; j += 256) {
        const int e    = j & 255;
        const int rest = j >> 8;                // b*64 + rank
        const int bb   = rest >> 6;
        const int r    = rest & 63;
        const int sa   = ord[(bb << 6) + r];        // tensor 0 groups 0..3
        const int sb2  = ord[((4 + bb) << 6) + r];  // tensor 1 groups 4..7
        const float av = embs[((bb)      * S_STEPS + sa ) * EMB_DIM + e];
        const float bv = embs[((4 + bb)  * S_STEPS + sb2) * EMB_DIM + e];
        const float d  = av - bv;
        acc += d * d;
    }
    red[tid] = acc;
    __syncthreads();
    for (int st = 128; st > 0; st >>= 1) {
        if (tid < st) red[tid] += red[tid + st];
        __syncthreads();
    }
    if (tid == 0) out[0] = red[0] * (1.0f / (float)(BATCH * S_STEPS * EMB_DIM));
}

// ---------------- launcher ----------------
extern "C" void kernel_launch(void* const* d_in, const int* in_sizes, int n_in,
                              void* d_out, int out_size, void* d_ws, size_t ws_size,
                              hipStream_t stream) {
    (void)in_sizes; (void)n_in; (void)out_size; (void)ws_size;
    const float* a_fms = (const float*)d_in[0];   // (64,4,256,1024) f32
    const float* b_fms = (const float*)d_in[1];   // (64,4,256,1024) f32
    const float* table = (const float*)d_in[2];   // (256,190) f32
    const float* w     = (const float*)d_in[3];   // (256,) f32

    char* ws = (char*)d_ws;
    float* pvals  = (float*)(ws);                                   // 2048 f32
    int*   pidxs  = (int*)  (ws + 2048 * 4);                        // 2048 i32
    float* embs   = (float*)(ws + 2048 * 8);                        // 512*256 f32
    float* scores = (float*)(ws + 2048 * 8 + (size_t)N_JOBS * EMB_DIM * 4); // 512 f32

    phase1_argmax<<<N_JOBS * PARTS, P1_THREADS, 0, stream>>>(a_fms, b_fms, pvals, pidxs);
    phase2_embed <<<N_JOBS, EMB_DIM, 0, stream>>>(pvals, pidxs, table, w, embs, scores);
    phase3_loss  <<<1, 256, 0, stream>>>(embs, scores, (float*)d_out);
}